// TemporalGNN_9809705304183
// MI455X (gfx1250) — compile-verified
//
#include <hip/hip_runtime.h>
#include <hip/hip_bf16.h>
#include <math.h>

// Problem dims (fixed by the reference)
#define BB   8
#define TT   6
#define NN   50000
#define EE   1600000
#define HID  16
#define TILES (NN / 16)          // 3125 16-node tiles per batch (divides exactly)
#define WPB  8                   // waves per block in the WMMA kernel

typedef float v2f __attribute__((ext_vector_type(2)));
typedef float v8f __attribute__((ext_vector_type(8)));

// ---- Kernel 1: degree via float atomics --------------------------------
__global__ void k_deg(const int* __restrict__ ei, float* __restrict__ deg) {
    int e = blockIdx.x * blockDim.x + threadIdx.x;
    if (e < EE) atomicAdd(&deg[ei[e]], 1.0f);          // ei[0:E] = src
}

// ---- Kernel 2: deg -> dinv in place ------------------------------------
__global__ void k_dinv(float* __restrict__ deg) {
    int n = blockIdx.x * blockDim.x + threadIdx.x;
    if (n < NN) {
        float d = deg[n];
        deg[n] = (d > 0.0f) ? rsqrtf(d) : 0.0f;
    }
}

// ---- Kernel 3: batched Laplacian scatter (only t = T-1 is live) --------
__global__ void k_scatter(const int* __restrict__ ei,
                          const float* __restrict__ x,
                          const float* __restrict__ dinv,
                          float* __restrict__ lx) {
    int e = blockIdx.x * blockDim.x + threadIdx.x;
    if (e >= EE) return;
    int s = ei[e];
    int t = ei[EE + e];
    float we = -dinv[s] * dinv[t];
#pragma unroll
    for (int b = 0; b < BB; ++b) {
        // x is [B, T, N, 1]; slice t = T-1
        float xv = x[((size_t)b * TT + (TT - 1)) * NN + s];
        atomicAdd(&lx[(size_t)b * NN + t], xv * we);
    }
}

// Branch-free (1 - sigmoid(a)) * tanh(b):
//   = e^{-a} (1 - e^{-2b}) / ((1+e^{-a})(1+e^{-2b}))
// using native v_exp_f32 / v_rcp_f32; inputs clamped to keep exp finite.
__device__ __forceinline__ float gate_fuse(float a, float b) {
    const float L2E = 1.4426950408889634f;
    a = fminf(fmaxf(a, -30.0f), 30.0f);
    b = fminf(fmaxf(b, -15.0f), 15.0f);
    float ez = __builtin_amdgcn_exp2f(-a * L2E);          // e^{-a}
    float eh = __builtin_amdgcn_exp2f(-2.0f * b * L2E);   // e^{-2b}
    float r  = __builtin_amdgcn_rcpf((1.0f + ez) * (1.0f + eh));
    return ez * (1.0f - eh) * r;
}

// ---- Kernel 4: fused GRU gates via V_WMMA_F32_16X16X4_F32 --------------
// D = A(16x4) x B(4x16):  A = [xf, lx, 1, 0] per node,  B = [w0; w1; bias; 0]
// A layout (32-bit, 16x4): lanes 0-15 -> K=0,1 in vgpr0,1 ; lanes 16-31 -> K=2,3
// B layout mirrors A; D: hid = lane%16, node = tile*16 + v + (lane>=16 ? 8 : 0)
__global__ void k_gru_wmma(const float* __restrict__ x,
                           const float* __restrict__ lx,
                           const float* __restrict__ w_xz,
                           const float* __restrict__ b_xz,
                           const float* __restrict__ b_hz,
                           const float* __restrict__ w_xh,
                           const float* __restrict__ b_xh,
                           const float* __restrict__ b_hh,
                           float* __restrict__ gsum) {
    __shared__ float lsum[HID];
    const int b    = blockIdx.y;
    const int lane = threadIdx.x & 31;
    const int wave = threadIdx.x >> 5;

    if (threadIdx.x < HID) lsum[threadIdx.x] = 0.0f;
    __syncthreads();

    const int tile = blockIdx.x * WPB + wave;       // wave-uniform guard:
    if (tile < TILES) {                             // EXEC stays all-1s at WMMA
        const float* xb  = x  + ((size_t)b * TT + (TT - 1)) * NN;
        const float* lxb = lx + (size_t)b * NN;
        const int node = tile * 16 + (lane & 15);

        v2f A, Bz, Bh;
        if (lane < 16) {
            A.x  = xb[node];            // K=0 : Tx0
            A.y  = lxb[node];           // K=1 : Tx1
            Bz.x = w_xz[lane];          // row K=0 : w0[h]
            Bz.y = w_xz[HID + lane];    // row K=1 : w1[h]
            Bh.x = w_xh[lane];
            Bh.y = w_xh[HID + lane];
        } else {
            const int h = lane - 16;
            A.x  = 1.0f;                // K=2 : bias selector
            A.y  = 0.0f;                // K=3 : zero
            Bz.x = b_xz[h] + b_hz[h];   // row K=2 : fused bias (x-path + h-path)
            Bz.y = 0.0f;
            Bh.x = b_xh[h] + b_hh[h];
            Bh.y = 0.0f;
        }

        v8f zero = {0.f, 0.f, 0.f, 0.f, 0.f, 0.f, 0.f, 0.f};
        v8f dz = __builtin_amdgcn_wmma_f32_16x16x4_f32(
            false, A, false, Bz, (short)0, zero, false, false);
        v8f dh = __builtin_amdgcn_wmma_f32_16x16x4_f32(
            false, A, false, Bh, (short)0, zero, false, false);

        // h = (1 - sigmoid(dz)) * tanh(dh); reduce 8 nodes per lane locally
        float acc = 0.0f;
#pragma unroll
        for (int v = 0; v < 8; ++v)
            acc += gate_fuse(dz[v], dh[v]);

        atomicAdd(&lsum[lane & 15], acc);           // ds_add_f32, hid-indexed
    }
    __syncthreads();

    if (threadIdx.x < HID)
        atomicAdd(&gsum[b * HID + threadIdx.x], lsum[threadIdx.x]);
}

// ---- Kernel 5: mean, ReLU, final linear --------------------------------
__global__ void k_final(const float* __restrict__ gsum,
                        const float* __restrict__ w_lin,
                        const float* __restrict__ b_lin,
                        float* __restrict__ out) {
    int b = threadIdx.x;
    if (b < BB) {
        float s = 0.0f;
#pragma unroll
        for (int h = 0; h < HID; ++h) {
            float m = gsum[b * HID + h] * (1.0f / (float)NN);
            s += fmaxf(m, 0.0f) * w_lin[h];
        }
        out[b] = s + b_lin[0];
    }
}

extern "C" void kernel_launch(void* const* d_in, const int* in_sizes, int n_in,
                              void* d_out, int out_size, void* d_ws, size_t ws_size,
                              hipStream_t stream) {
    const float* x     = (const float*)d_in[0];
    const int*   ei    = (const int*)  d_in[1];
    const float* w_xz  = (const float*)d_in[2];
    const float* b_xz  = (const float*)d_in[3];
    const float* b_hz  = (const float*)d_in[4];
    const float* w_xh  = (const float*)d_in[5];
    const float* b_xh  = (const float*)d_in[6];
    const float* b_hh  = (const float*)d_in[7];
    const float* w_lin = (const float*)d_in[8];
    const float* b_lin = (const float*)d_in[9];
    float* out = (float*)d_out;

    // Workspace layout: [dinv: N][lx: B*N][gsum: B*HID]  (~1.8 MB)
    float* ws   = (float*)d_ws;
    float* deg  = ws;                       // becomes dinv in place
    float* lx   = ws + NN;
    float* gsum = ws + NN + (size_t)BB * NN;
    size_t zbytes = (size_t)(NN + (size_t)BB * NN + BB * HID) * sizeof(float);
    hipMemsetAsync(d_ws, 0, zbytes, stream);

    k_deg    <<<(EE + 255) / 256, 256, 0, stream>>>(ei, deg);
    k_dinv   <<<(NN + 255) / 256, 256, 0, stream>>>(deg);
    k_scatter<<<(EE + 255) / 256, 256, 0, stream>>>(ei, x, deg, lx);

    dim3 grid((TILES + WPB - 1) / WPB, BB);
    k_gru_wmma<<<grid, 32 * WPB, 0, stream>>>(x, lx, w_xz, b_xz, b_hz,
                                              w_xh, b_xh, b_hh, gsum);

    k_final<<<1, 32, 0, stream>>>(gsum, w_lin, b_lin, out);
}